// DecRnn_21285857919004
// MI455X (gfx1250) — compile-verified
//
#include <hip/hip_runtime.h>
#include <math.h>

typedef __attribute__((ext_vector_type(2))) float v2f;
typedef __attribute__((ext_vector_type(8))) float v8f;

constexpr int H  = 512;
constexpr int E  = 300;
constexpr int V  = 50257;
constexpr int B  = 64;
constexpr int S  = 128;
constexpr int H2 = 2 * H;          // 1024
constexpr int H3 = 3 * H;          // 1536
constexpr int KX = E + H2;         // 1324  (GRU input concat)
constexpr int KY = H3 + E;         // 1836  (output concat: h_new | weighted | embedded)
constexpr float NEGV = -1e10f;

// D = A(16x4,f32) * B(4x16,f32) + C   — CDNA5 FP32 WMMA
__device__ __forceinline__ v8f wmma4(v2f a, v2f b, v8f c) {
  return __builtin_amdgcn_wmma_f32_16x16x4_f32(false, a, false, b, (short)0, c,
                                               false, false);
}

// Fragment load: for both A (16x4, row m=l) and B (4x16, col n=l), lane
// (half,l) needs the two consecutive K values {k0+2*half, k0+2*half+1} of its
// row/column -> single global_load_b64, no lane-dependent selects.
__device__ __forceinline__ v2f frag(const float* p, int k0) {
  return *(const v2f*)(p + k0);
}

// ---------------------------------------------------------------------------
// K0: embedding gather -> xbuf[:, 0:E] and ybuf[:, H3:H3+E]
// ---------------------------------------------------------------------------
__global__ void embed_gather(const int* __restrict__ ids,
                             const float* __restrict__ table,
                             float* __restrict__ xbuf, float* __restrict__ ybuf) {
  const int b  = blockIdx.x;
  const int id = ids[b];
  for (int e = threadIdx.x; e < E; e += blockDim.x) {
    float v = table[(size_t)id * E + e];
    xbuf[b * KX + e]      = v;
    ybuf[b * KY + H3 + e] = v;
  }
}

// ---------------------------------------------------------------------------
// Generic wave-tile GEMM:  C[m,n] = sum_k A[m,k] * W[n,k] + bias[n]
// A: [M, K] lda ; W: [N, K] ldw (K contiguous) ; one wave per 16x16 tile.
// grid = (N/16, M/16), block = 32. K % 4 == 0.
// ---------------------------------------------------------------------------
__global__ __launch_bounds__(32)
void gemm_atb(const float* __restrict__ A, int lda,
              const float* __restrict__ W, int ldw,
              const float* __restrict__ bias,
              float* __restrict__ C, int ldc, int K) {
  const int lane = threadIdx.x;
  const int half = lane >> 4;        // 0: K={0,1}, 1: K={2,3}
  const int l    = lane & 15;
  const int n0   = blockIdx.x * 16;
  const int m0   = blockIdx.y * 16;

  v8f c = {};
  const float* Ar = A + (size_t)(m0 + l) * lda + 2 * half;
  const float* Wr = W + (size_t)(n0 + l) * ldw + 2 * half;

  const int Kmain = K & ~15;
  int k0 = 0;
  for (; k0 < Kmain; k0 += 16) {     // 8 b64 loads clause, then 4 WMMAs
    v2f a0 = frag(Ar, k0 + 0), a1 = frag(Ar, k0 + 4);
    v2f a2 = frag(Ar, k0 + 8), a3 = frag(Ar, k0 + 12);
    v2f b0 = frag(Wr, k0 + 0), b1 = frag(Wr, k0 + 4);
    v2f b2 = frag(Wr, k0 + 8), b3 = frag(Wr, k0 + 12);
    c = wmma4(a0, b0, c);
    c = wmma4(a1, b1, c);
    c = wmma4(a2, b2, c);
    c = wmma4(a3, b3, c);
  }
  for (; k0 < K; k0 += 4) {          // K % 16 remainder (12 for KX)
    c = wmma4(frag(Ar, k0), frag(Wr, k0), c);
  }

  const int n = n0 + l;
  const float bn = bias ? bias[n] : 0.f;
  for (int i = 0; i < 8; ++i) {
    int m = m0 + i + 8 * half;       // C layout: VGPR i -> M=i (lanes 0-15), M=i+8 (16-31)
    C[(size_t)m * ldc + n] = c[i] + bn;
  }
}

// ---------------------------------------------------------------------------
// K2: fused attention energy + tanh + v-dot + mask.
// Rows: sb = s*B + b (enc layout [S,B,2H]); 16-row tile => fixed s, 16 b's.
// 8 waves/block, each wave owns 4 N-tiles (covers N = H = 512).
// scores[b,s] = sum_h v[h]*tanh(ep[sb,h] + hp[b,h])  (hp already has b_attn)
// ---------------------------------------------------------------------------
__global__ __launch_bounds__(256)
void attn_scores(const float* __restrict__ enc,   // [S*B, 2H]
                 const float* __restrict__ Wep,   // W_attn + H, ld = 3H
                 const float* __restrict__ hp,    // [B, H]
                 const float* __restrict__ vw,    // [H]
                 const int* __restrict__ mask,    // [B, S]
                 float* __restrict__ scores) {    // [B, S]
  __shared__ float sc[16];
  const int tid  = threadIdx.x;
  const int wave = tid >> 5;
  const int lane = tid & 31;
  const int half = lane >> 4;
  const int l    = lane & 15;
  const int sb0  = blockIdx.x * 16;
  const int s    = sb0 >> 6;         // sb0 / B
  const int b0   = sb0 & 63;         // sb0 % B
  if (tid < 16) sc[tid] = 0.f;
  __syncthreads();

  v8f c0 = {}, c1 = {}, c2 = {}, c3 = {};
  const float* Ar = enc + (size_t)(sb0 + l) * H2 + 2 * half;
  const int nbase = wave * 64;
  const float* W0 = Wep + (size_t)(nbase +  0 + l) * H3 + 2 * half;
  const float* W1 = Wep + (size_t)(nbase + 16 + l) * H3 + 2 * half;
  const float* W2 = Wep + (size_t)(nbase + 32 + l) * H3 + 2 * half;
  const float* W3 = Wep + (size_t)(nbase + 48 + l) * H3 + 2 * half;
  for (int k0 = 0; k0 < H2; k0 += 16) {   // 20 b64 loads, then 16 WMMAs
    v2f a0 = frag(Ar, k0 + 0), a1 = frag(Ar, k0 + 4);
    v2f a2 = frag(Ar, k0 + 8), a3 = frag(Ar, k0 + 12);
    v2f p00 = frag(W0, k0 + 0), p01 = frag(W0, k0 + 4);
    v2f p02 = frag(W0, k0 + 8), p03 = frag(W0, k0 + 12);
    v2f p10 = frag(W1, k0 + 0), p11 = frag(W1, k0 + 4);
    v2f p12 = frag(W1, k0 + 8), p13 = frag(W1, k0 + 12);
    v2f p20 = frag(W2, k0 + 0), p21 = frag(W2, k0 + 4);
    v2f p22 = frag(W2, k0 + 8), p23 = frag(W2, k0 + 12);
    v2f p30 = frag(W3, k0 + 0), p31 = frag(W3, k0 + 4);
    v2f p32 = frag(W3, k0 + 8), p33 = frag(W3, k0 + 12);
    c0 = wmma4(a0, p00, c0); c1 = wmma4(a0, p10, c1);
    c2 = wmma4(a0, p20, c2); c3 = wmma4(a0, p30, c3);
    c0 = wmma4(a1, p01, c0); c1 = wmma4(a1, p11, c1);
    c2 = wmma4(a1, p21, c2); c3 = wmma4(a1, p31, c3);
    c0 = wmma4(a2, p02, c0); c1 = wmma4(a2, p12, c1);
    c2 = wmma4(a2, p22, c2); c3 = wmma4(a2, p32, c3);
    c0 = wmma4(a3, p03, c0); c1 = wmma4(a3, p13, c1);
    c2 = wmma4(a3, p23, c2); c3 = wmma4(a3, p33, c3);
  }

  v8f acc[4] = { c0, c1, c2, c3 };
  for (int t = 0; t < 4; ++t) {
    const int h  = nbase + t * 16 + l;
    const float vh = vw[h];
    for (int i = 0; i < 8; ++i) {
      const int b = b0 + i + 8 * half;
      float p = vh * tanhf(acc[t][i] + hp[b * H + h]);
      // reduce over the 16 lanes of each half-wave (N dimension)
      p += __shfl_xor(p, 1, 32);
      p += __shfl_xor(p, 2, 32);
      p += __shfl_xor(p, 4, 32);
      p += __shfl_xor(p, 8, 32);
      if (l == 0) atomicAdd(&sc[i + 8 * half], p);   // ds_add_f32
    }
  }
  __syncthreads();
  if (tid < 16) {
    const int b = b0 + tid;
    scores[b * S + s] = (mask[b * S + s] == 0) ? NEGV : sc[tid];
  }
}

// ---------------------------------------------------------------------------
// K3: softmax over S per batch row + context vector; writes 'a' output and
// weighted into xbuf[:, E:E+2H] / ybuf[:, H:H+2H].
// ---------------------------------------------------------------------------
__global__ __launch_bounds__(256)
void softmax_weighted(const float* __restrict__ scores,
                      const float* __restrict__ enc,
                      float* __restrict__ a_out,
                      float* __restrict__ xbuf, float* __restrict__ ybuf) {
  __shared__ float sa[S];
  __shared__ float red;
  const int b   = blockIdx.x;
  const int tid = threadIdx.x;
  if (tid < S) sa[tid] = scores[b * S + tid];
  __syncthreads();
  if (tid == 0) {
    float mx = sa[0];
    for (int i = 1; i < S; ++i) mx = fmaxf(mx, sa[i]);
    red = mx;
  }
  __syncthreads();
  const float mx = red;
  if (tid < S) sa[tid] = expf(sa[tid] - mx);
  __syncthreads();
  if (tid == 0) {
    float sm = 0.f;
    for (int i = 0; i < S; ++i) sm += sa[i];
    red = 1.f / sm;
  }
  __syncthreads();
  const float inv = red;
  if (tid < S) {
    float av = sa[tid] * inv;
    sa[tid] = av;
    a_out[b * S + tid] = av;
  }
  __syncthreads();
  for (int e2 = tid; e2 < H2; e2 += 256) {
    float acc = 0.f;
    for (int s = 0; s < S; ++s)
      acc += sa[s] * enc[(size_t)(s * B + b) * H2 + e2];
    xbuf[b * KX + E + e2] = acc;
    ybuf[b * KY + H + e2] = acc;
  }
}

// ---------------------------------------------------------------------------
// K6: GRU gate combine (PyTorch order r,z,n); writes h_new output and
// ybuf[:, 0:H].
// ---------------------------------------------------------------------------
__global__ __launch_bounds__(256)
void gru_combine(const float* __restrict__ gx, const float* __restrict__ gh,
                 const float* __restrict__ hidden,
                 float* __restrict__ hnew_out, float* __restrict__ ybuf) {
  const int idx = blockIdx.x * blockDim.x + threadIdx.x;
  if (idx >= B * H) return;
  const int b = idx / H, h = idx % H;
  const float* gxr = gx + b * H3;
  const float* ghr = gh + b * H3;
  const float r = 1.f / (1.f + expf(-(gxr[h] + ghr[h])));
  const float z = 1.f / (1.f + expf(-(gxr[H + h] + ghr[H + h])));
  const float n = tanhf(gxr[2 * H + h] + r * ghr[2 * H + h]);
  const float hn = (1.f - z) * n + z * hidden[idx];
  hnew_out[idx]    = hn;
  ybuf[b * KY + h] = hn;
}

// ---------------------------------------------------------------------------
// K5: vocab projection. HBM-bound on W_out (369 MB) -> read it exactly once:
// each wave owns one 16-wide vocab slab and ALL 4 M-tiles of B=64.
// pred[b,v] = sum_k y[b,k] * W_out[v,k] + b_out[v]
// ---------------------------------------------------------------------------
__global__ __launch_bounds__(32)
void out_proj(const float* __restrict__ Y,     // [B, KY]
              const float* __restrict__ Wout,  // [V, KY]
              const float* __restrict__ bout,  // [V]
              float* __restrict__ pred) {      // [B, V]
  const int lane = threadIdx.x;
  const int half = lane >> 4;
  const int l    = lane & 15;
  const int n0   = blockIdx.x * 16;
  const int n    = n0 + l;
  const int nc   = (n < V) ? n : (V - 1);      // clamp reads, guard stores

  v8f c0 = {}, c1 = {}, c2 = {}, c3 = {};
  const float* A0 = Y + (size_t)( 0 + l) * KY + 2 * half;
  const float* A1 = Y + (size_t)(16 + l) * KY + 2 * half;
  const float* A2 = Y + (size_t)(32 + l) * KY + 2 * half;
  const float* A3 = Y + (size_t)(48 + l) * KY + 2 * half;
  const float* Wr = Wout + (size_t)nc * KY + 2 * half;

  const int Kmain = KY & ~15;                  // 1824; remainder 12
  int k0 = 0;
  for (; k0 < Kmain; k0 += 16) {               // 20 b64 loads, then 16 WMMAs
    __builtin_prefetch(Wr + k0 + 512, 0, 1);   // global_prefetch_b8: W_out stream
    v2f b0 = frag(Wr, k0 + 0), b1 = frag(Wr, k0 + 4);
    v2f b2 = frag(Wr, k0 + 8), b3 = frag(Wr, k0 + 12);
    v2f a00 = frag(A0, k0 + 0), a01 = frag(A0, k0 + 4);
    v2f a02 = frag(A0, k0 + 8), a03 = frag(A0, k0 + 12);
    v2f a10 = frag(A1, k0 + 0), a11 = frag(A1, k0 + 4);
    v2f a12 = frag(A1, k0 + 8), a13 = frag(A1, k0 + 12);
    v2f a20 = frag(A2, k0 + 0), a21 = frag(A2, k0 + 4);
    v2f a22 = frag(A2, k0 + 8), a23 = frag(A2, k0 + 12);
    v2f a30 = frag(A3, k0 + 0), a31 = frag(A3, k0 + 4);
    v2f a32 = frag(A3, k0 + 8), a33 = frag(A3, k0 + 12);
    c0 = wmma4(a00, b0, c0); c1 = wmma4(a10, b0, c1);
    c2 = wmma4(a20, b0, c2); c3 = wmma4(a30, b0, c3);
    c0 = wmma4(a01, b1, c0); c1 = wmma4(a11, b1, c1);
    c2 = wmma4(a21, b1, c2); c3 = wmma4(a31, b1, c3);
    c0 = wmma4(a02, b2, c0); c1 = wmma4(a12, b2, c1);
    c2 = wmma4(a22, b2, c2); c3 = wmma4(a32, b2, c3);
    c0 = wmma4(a03, b3, c0); c1 = wmma4(a13, b3, c1);
    c2 = wmma4(a23, b3, c2); c3 = wmma4(a33, b3, c3);
  }
  for (; k0 < KY; k0 += 4) {
    v2f b = frag(Wr, k0);
    c0 = wmma4(frag(A0, k0), b, c0);
    c1 = wmma4(frag(A1, k0), b, c1);
    c2 = wmma4(frag(A2, k0), b, c2);
    c3 = wmma4(frag(A3, k0), b, c3);
  }

  if (n < V) {
    const float bn = bout[n];
    for (int i = 0; i < 8; ++i) {
      const int m = i + 8 * half;
      pred[(size_t)( 0 + m) * V + n] = c0[i] + bn;
      pred[(size_t)(16 + m) * V + n] = c1[i] + bn;
      pred[(size_t)(32 + m) * V + n] = c2[i] + bn;
      pred[(size_t)(48 + m) * V + n] = c3[i] + bn;
    }
  }
}

// ---------------------------------------------------------------------------
extern "C" void kernel_launch(void* const* d_in, const int* in_sizes, int n_in,
                              void* d_out, int out_size, void* d_ws, size_t ws_size,
                              hipStream_t stream) {
  (void)in_sizes; (void)n_in; (void)out_size; (void)ws_size;

  const int*   ids    = (const int*)  d_in[0];
  const float* hidden = (const float*)d_in[1];
  const float* enc    = (const float*)d_in[2];   // [S, B, 2H]
  const int*   mask   = (const int*)  d_in[3];
  const float* emb    = (const float*)d_in[4];
  const float* W_attn = (const float*)d_in[5];   // [H, 3H]
  const float* b_attn = (const float*)d_in[6];
  const float* v_w    = (const float*)d_in[7];
  const float* W_ih   = (const float*)d_in[8];   // [3H, KX]
  const float* W_hh   = (const float*)d_in[9];   // [3H, H]
  const float* b_ih   = (const float*)d_in[10];
  const float* b_hh   = (const float*)d_in[11];
  const float* W_out  = (const float*)d_in[12];  // [V, KY]
  const float* b_out  = (const float*)d_in[13];

  float* out   = (float*)d_out;
  float* pred  = out;                        // [B, V]
  float* hnew  = out + (size_t)B * V;        // [B, H]
  float* a_out = hnew + (size_t)B * H;       // [B, S]

  float* ws   = (float*)d_ws;
  float* hp   = ws;                          // [B, H]   hidden @ W_attn[:,:H].T + b_attn
  float* scr  = hp  + B * H;                 // [B, S]   masked scores
  float* xbuf = scr + B * S;                 // [B, KX]  [embedded | weighted]
  float* ybuf = xbuf + B * KX;               // [B, KY]  [h_new | weighted | embedded]
  float* gx   = ybuf + B * KY;               // [B, 3H]
  float* gh   = gx   + B * H3;               // [B, 3H]

  embed_gather<<<B, 256, 0, stream>>>(ids, emb, xbuf, ybuf);

  // hp[b,h] = hidden @ W_attn[:, :H].T + b_attn   (S-invariant half of energy)
  gemm_atb<<<dim3(H / 16, B / 16), 32, 0, stream>>>(hidden, H, W_attn, H3,
                                                    b_attn, hp, H, H);

  attn_scores<<<(S * B) / 16, 256, 0, stream>>>(enc, W_attn + H, hp, v_w, mask, scr);

  softmax_weighted<<<B, 256, 0, stream>>>(scr, enc, a_out, xbuf, ybuf);

  // gx = x @ W_ih.T + b_ih ; gh = hidden @ W_hh.T + b_hh
  gemm_atb<<<dim3(H3 / 16, B / 16), 32, 0, stream>>>(xbuf, KX, W_ih, KX,
                                                     b_ih, gx, H3, KX);
  gemm_atb<<<dim3(H3 / 16, B / 16), 32, 0, stream>>>(hidden, H, W_hh, H,
                                                     b_hh, gh, H3, H);

  gru_combine<<<(B * H) / 256, 256, 0, stream>>>(gx, gh, hidden, hnew, ybuf);

  out_proj<<<(V + 15) / 16, 32, 0, stream>>>(ybuf, W_out, b_out, pred);
}